// TAA_38182259261484
// MI455X (gfx1250) — compile-verified
//
#include <hip/hip_runtime.h>
#include <cstdint>

// TAA resolve for MI455X (gfx1250). Bandwidth-bound: ~183 MB unique traffic
// @ 23.3 TB/s ≈ 8 us floor. CDNA5 paths: per-lane async global->LDS gather
// (ASYNCcnt) with NT cache hint + s_wait_asynccnt split counter. Single-use
// streams (x, mv, out) are non-temporal so the reused history stream (50 MB)
// stays resident in the 192 MB L2.

namespace {
constexpr int N_ = 2, C_ = 3, H_ = 1080, W_ = 1920;
constexpr int BX = 32, BY = 8;             // 256 threads = 8 wave32
constexpr int TW = BX + 2, THALO = BY + 2; // 34 x 10 halo tile
constexpr int TILE = TW * THALO;           // 340 floats / channel
constexpr float ALPHA = 0.1f;
}

typedef float v2f __attribute__((ext_vector_type(2)));

__device__ __forceinline__ int iclamp(int v, int lo, int hi) {
    return v < lo ? lo : (v > hi ? hi : v);
}

// gfx1250 async copy: copies the float at g to LDS byte address lds_byte (per active lane), tracked by
// ASYNCcnt. NT hint: x is read exactly once, keep it out of L2's hot set.
__device__ __forceinline__ void async_ld_b32_nt(uint32_t lds_byte, const float* g) {
    asm volatile("global_load_async_to_lds_b32 %0, %1, off th:TH_LOAD_NT"
                 :: "v"(lds_byte), "v"(g) : "memory");
}

struct Axis { int i0, i1; float w; };

// Bilinear axis prep for border-clamped sampling at pixel-space coordinate p.
__device__ __forceinline__ Axis axis_prep(float p, int n) {
    float f0 = floorf(p);
    Axis a;
    a.w = p - f0;
    int i = (int)f0;
    a.i0 = iclamp(i, 0, n - 1);
    a.i1 = iclamp(i + 1, 0, n - 1);
    return a;
}

__device__ __forceinline__ float bilin(const float* __restrict__ h,
                                       int r0, int r1, const Axis& X, float wy) {
    float v00 = h[r0 + X.i0];
    float v01 = h[r0 + X.i1];
    float v10 = h[r1 + X.i0];
    float v11 = h[r1 + X.i1];
    float top = v00 * (1.0f - X.w) + v01 * X.w;
    float bot = v10 * (1.0f - X.w) + v11 * X.w;
    return top * (1.0f - wy) + bot * wy;
}

__global__ __launch_bounds__(BX * BY)
void taa_resolve_kernel(const float* __restrict__ x,
                        const float* __restrict__ mv,
                        const float* __restrict__ hist,
                        float* __restrict__ out) {
    __shared__ float tile[C_ * TILE];

    const int n   = blockIdx.z;
    const int bx0 = blockIdx.x * BX;
    const int by0 = blockIdx.y * BY;
    const int lx  = threadIdx.x;
    const int ly  = threadIdx.y;
    const int tid = ly * BX + lx;

    // ---- Stage x halo tile (3 channels, clamp-to-edge) into LDS via async
    // copies. Clamp duplication == SAME/-inf padding for min & max pooling.
    const uint32_t lds_base = (uint32_t)(uintptr_t)(&tile[0]);
#pragma unroll
    for (int c = 0; c < C_; ++c) {
        const float* __restrict__ xc_plane = x + ((size_t)(n * C_ + c) * H_) * W_;
        const uint32_t lds_c = lds_base + (uint32_t)(c * TILE) * 4u;
#pragma unroll
        for (int k = 0; k < 2; ++k) {
            int i = tid + k * (BX * BY);
            if (i < TILE) {
                int r   = i / TW;          // /34: constant mul-shift
                int col = i - r * TW;
                int gy  = iclamp(by0 + r - 1, 0, H_ - 1);
                int gx  = iclamp(bx0 + col - 1, 0, W_ - 1);
                async_ld_b32_nt(lds_c + (uint32_t)i * 4u,
                                xc_plane + ((size_t)gy * W_ + gx));
            }
        }
    }
    asm volatile("s_wait_asynccnt 0x0" ::: "memory");
    __syncthreads();

    // ---- Per-pixel bicubic (Catmull-Rom, 5-tap) setup, shared by channels.
    const int px = bx0 + lx;
    const int py = by0 + ly;
    const size_t pix = ((size_t)n * H_ + py) * W_ + px;
    const v2f g = __builtin_nontemporal_load(
        reinterpret_cast<const v2f*>(mv) + pix);

    const float posx = (g.x + 1.0f) * 0.5f * (float)W_;
    const float posy = (g.y + 1.0f) * 0.5f * (float)H_;
    const float cpx = floorf(posx - 0.5f) + 0.5f;
    const float cpy = floorf(posy - 0.5f) + 0.5f;
    const float fx = posx - cpx, fy = posy - cpy;
    const float fx2 = fx * fx, fx3 = fx2 * fx;
    const float fy2 = fy * fy, fy3 = fy2 * fy;

    const float w0x  = -0.5f * fx3 + fx2 - 0.5f * fx;
    const float w2x  = -1.5f * fx3 + 2.0f * fx2 + 0.5f * fx;
    const float w3x  =  0.5f * fx3 - 0.5f * fx2;
    const float w12x = (1.5f * fx3 - 2.5f * fx2 + 1.0f) + w2x;
    const float w0y  = -0.5f * fy3 + fy2 - 0.5f * fy;
    const float w2y  = -1.5f * fy3 + 2.0f * fy2 + 0.5f * fy;
    const float w3y  =  0.5f * fy3 - 0.5f * fy2;
    const float w12y = (1.5f * fy3 - 2.5f * fy2 + 1.0f) + w2y;

    // Tap pixel-space coordinates (tc*Dim - 0.5 folded in).
    const Axis X0  = axis_prep(cpx - 1.5f,               W_);
    const Axis X12 = axis_prep(cpx + w2x / w12x - 0.5f,  W_);
    const Axis X3  = axis_prep(cpx + 1.5f,               W_);
    const Axis Y0  = axis_prep(cpy - 1.5f,               H_);
    const Axis Y12 = axis_prep(cpy + w2y / w12y - 0.5f,  H_);
    const Axis Y3  = axis_prep(cpy + 1.5f,               H_);
    const int r0_0  = Y0.i0  * W_, r1_0  = Y0.i1  * W_;
    const int r0_12 = Y12.i0 * W_, r1_12 = Y12.i1 * W_;
    const int r0_3  = Y3.i0  * W_, r1_3  = Y3.i1  * W_;

    const float ww1 = w12x * w0y;
    const float ww2 = w0x  * w12y;
    const float ww3 = w3x  * w12y;
    const float ww4 = w12x * w3y;
    const float ww5 = w12x * w12y;
    const float rrec = 1.0f / (ww1 + ww2 + ww3 + ww4 + ww5);

    // ---- Per channel: 5 bilinear taps + LDS 3x3 clamp + blend.
#pragma unroll
    for (int c = 0; c < C_; ++c) {
        const float* __restrict__ h = hist + ((size_t)(n * C_ + c) * H_) * W_;
        float s1 = bilin(h, r0_0,  r1_0,  X12, Y0.w);   // (tc12.x, tc0.y)
        float s2 = bilin(h, r0_12, r1_12, X0,  Y12.w);  // (tc0.x,  tc12.y)
        float s3 = bilin(h, r0_12, r1_12, X3,  Y12.w);  // (tc3.x,  tc12.y)
        float s4 = bilin(h, r0_3,  r1_3,  X12, Y3.w);   // (tc12.x, tc3.y)
        float s5 = bilin(h, r0_12, r1_12, X12, Y12.w);  // (tc12.x, tc12.y)
        float reproj = (s1 * ww1 + s2 * ww2 + s3 * ww3 + s4 * ww4 + s5 * ww5) * rrec;

        const float* tc = &tile[c * TILE + ly * TW + lx];
        float mx = -3.402823466e+38f;
        float mn =  3.402823466e+38f;
#pragma unroll
        for (int dy = 0; dy < 3; ++dy) {
            const float* row = tc + dy * TW;
#pragma unroll
            for (int dx = 0; dx < 3; ++dx) {
                float v = row[dx];
                mx = fmaxf(mx, v);
                mn = fminf(mn, v);
            }
        }
        float xc = tc[TW + 1];  // center pixel of x from LDS
        float rp = fminf(fmaxf(reproj, mn), mx);
        float result = ALPHA * xc + (1.0f - ALPHA) * rp;
        __builtin_nontemporal_store(
            result, &out[((size_t)(n * C_ + c) * H_ + py) * W_ + px]);
    }
}

extern "C" void kernel_launch(void* const* d_in, const int* in_sizes, int n_in,
                              void* d_out, int out_size, void* d_ws, size_t ws_size,
                              hipStream_t stream) {
    (void)in_sizes; (void)n_in; (void)out_size; (void)d_ws; (void)ws_size;
    const float* x    = (const float*)d_in[0];
    const float* mv   = (const float*)d_in[1];
    const float* hist = (const float*)d_in[2];
    float* out = (float*)d_out;

    dim3 block(BX, BY, 1);
    dim3 grid(W_ / BX, H_ / BY, N_);  // 60 x 135 x 2, exact tiling
    taa_resolve_kernel<<<grid, block, 0, stream>>>(x, mv, hist, out);
}